// Sinkhorn_seq_29824252903725
// MI455X (gfx1250) — compile-verified
//
#include <hip/hip_runtime.h>
#include <math.h>
#include <stdint.h>

// ---------------------------------------------------------------------------
// Sinkhorn EMD loss, B=4, N=256, D=512, f32 throughout.
// softmax -> L1 cost matrix (+K=exp(-Wc/eps)) -> LDS-resident Sinkhorn
// (TDM tensor_load_to_lds staging, WMMA f32 16x16x4 row/col reductions)
// -> scalar loss.
// ---------------------------------------------------------------------------

typedef __attribute__((ext_vector_type(2))) float v2f;
typedef __attribute__((ext_vector_type(8))) float v8f;
typedef __attribute__((ext_vector_type(4))) unsigned int v4u;
typedef __attribute__((ext_vector_type(8))) int v8i;
typedef __attribute__((ext_vector_type(4))) int v4i;

#define NB 4
#define NN 256
#define ND 512
#define PITCH 257      // LDS pitch for 256x256 P matrix (kills bank conflicts)
#define DK 128         // D-stage for cost matrix tiling
#define CPITCH 132     // padded pitch (multiple of 4 for float4, !=0 mod 64)

__device__ __forceinline__ float waveReduceSum(float v) {
    #pragma unroll
    for (int m = 16; m >= 1; m >>= 1) v += __shfl_xor(v, m, 32);
    return v;
}
__device__ __forceinline__ float waveReduceMax(float v) {
    #pragma unroll
    for (int m = 16; m >= 1; m >>= 1) v = fmaxf(v, __shfl_xor(v, m, 32));
    return v;
}

// ------------------------- softmax(y/2) per row ----------------------------
__global__ void softmax_rows(const float* __restrict__ y_s,
                             const float* __restrict__ y_t,
                             float* __restrict__ p_s,
                             float* __restrict__ p_t) {
    const int wave = threadIdx.x >> 5;
    const int lane = threadIdx.x & 31;
    const int row  = blockIdx.x * 8 + wave;          // 0..2047
    const float* src;
    float* dst;
    if (row < NB * NN) { src = y_s + row * ND;              dst = p_s + row * ND; }
    else               { src = y_t + (row - NB * NN) * ND;  dst = p_t + (row - NB * NN) * ND; }

    float z[16];
    float m = -3.0e38f;
    #pragma unroll
    for (int k = 0; k < 16; ++k) {
        z[k] = 0.5f * src[lane + 32 * k];            // y / T, T = 2
        m = fmaxf(m, z[k]);
    }
    m = waveReduceMax(m);
    float s = 0.f;
    #pragma unroll
    for (int k = 0; k < 16; ++k) { z[k] = __expf(z[k] - m); s += z[k]; }
    s = waveReduceSum(s);
    const float inv = 1.0f / s;
    #pragma unroll
    for (int k = 0; k < 16; ++k) dst[lane + 32 * k] = z[k] * inv;
}

// ---------------- L1 cost matrix + Gibbs kernel K --------------------------
__global__ void cost_matrix(const float* __restrict__ p_s,
                            const float* __restrict__ p_t,
                            float* __restrict__ Wg,
                            float* __restrict__ Kg) {
    __shared__ __align__(16) float As[16 * CPITCH];
    __shared__ __align__(16) float Bs[16 * CPITCH];
    const int tx = threadIdx.x & 15;   // j within tile
    const int ty = threadIdx.x >> 4;   // i within tile
    const int b  = blockIdx.z;
    const int i0 = blockIdx.y * 16;
    const int j0 = blockIdx.x * 16;
    const float* Sa = p_s + (b * NN + i0) * ND;
    const float* Sb = p_t + (b * NN + j0) * ND;

    float acc = 0.f;
    for (int d0 = 0; d0 < ND; d0 += DK) {
        __syncthreads();
        for (int t = threadIdx.x; t < 16 * DK; t += 256) {
            const int r = t >> 7, c = t & (DK - 1);
            As[r * CPITCH + c] = Sa[r * ND + d0 + c];
            Bs[r * CPITCH + c] = Sb[r * ND + d0 + c];
        }
        if (d0 + DK < ND) {                      // global_prefetch_b8 next stage
            __builtin_prefetch(&Sa[ty * ND + d0 + DK], 0, 1);
            __builtin_prefetch(&Sb[ty * ND + d0 + DK], 0, 1);
        }
        __syncthreads();
        const float4* a4 = (const float4*)&As[ty * CPITCH];
        const float4* b4 = (const float4*)&Bs[tx * CPITCH];
        #pragma unroll
        for (int d = 0; d < DK / 4; ++d) {
            const float4 a = a4[d], q = b4[d];
            acc += fabsf(a.x - q.x) + fabsf(a.y - q.y) +
                   fabsf(a.z - q.z) + fabsf(a.w - q.w);
        }
    }
    const int idx = (b * NN + (i0 + ty)) * NN + (j0 + tx);
    Wg[idx] = acc;
    const float wc = fminf(acc, 10.0f);
    Kg[idx] = __expf(-wc * 10.0f) + 1e-8f;       // exp(-Wc/0.1) + 1e-8
}

// ----------------- LDS-resident Sinkhorn, TDM + WMMA -----------------------
// One block per batch, 512 threads (16 waves). P (256x256, pitch 257) lives
// entirely in the 320KB WGP LDS. K is DMA'd in by the Tensor Data Mover with
// hardware row padding (1 DWORD per 256-DWORD row => pitch 257). Row/col
// sums via V_WMMA_F32_16X16X4_F32.
__global__ void sinkhorn_lds(const float* __restrict__ Wg,
                             const float* __restrict__ Kg,
                             float* __restrict__ losses) {
    extern __shared__ float smem[];
    float* P  = smem;                   // NN * PITCH
    float* rs = smem + NN * PITCH;      // NN row sums -> reciprocals
    float* cs = rs + NN;                // NN col sums -> reciprocals
    const int b    = blockIdx.x;
    const int tid  = threadIdx.x;       // 0..511
    const int lane = tid & 31;
    const int wave = tid >> 5;          // 0..15
    const float* Kb = Kg + b * NN * NN;
    const float* Wb = Wg + b * NN * NN;

#if defined(__has_builtin) && __has_builtin(__builtin_amdgcn_tensor_load_to_lds)
    if (wave == 0) {
        // --- Tensor DMA descriptor (D#): 2D 256x256 f32 tile, padded rows ---
        const unsigned long long ga = (unsigned long long)(uintptr_t)Kb;
        v4u g0;
        g0[0] = 1u;                                        // count=1 (valid)
        g0[1] = 0u;                                        // lds_addr = 0 (dyn LDS base)
        g0[2] = (unsigned)(ga & 0xFFFFFFFFull);            // global_addr[31:0]
        g0[3] = (unsigned)((ga >> 32) & 0x1FFFFFFull)      // global_addr[56:32]
              | (2u << 30);                                // type = 2 ("image")
        v8i g1;
        g1[0] = (2 << 16)      // data_size = 4B
              | (1 << 20)      // pad_enable
              | (7 << 22)      // pad_interval: 256 DWORDs (= one 256-elem row)
              | (0 << 25);     // pad_amount: 1 DWORD  => LDS pitch 257
        g1[1] = (int)(256u << 16);   // tensor_dim0[15:0] in [31:16]
        g1[2] = (int)(256u << 16);   // tensor_dim0 hi=0; tensor_dim1[15:0]
        g1[3] = (int)(256u << 16);   // tensor_dim1 hi=0; tile_dim0 = 256
        g1[4] = 256;                 // tile_dim1 = 256; tile_dim2 = 0
        g1[5] = 256;                 // tensor_dim0_stride = 256 elements
        g1[6] = 0;
        g1[7] = 0;
        const v4i gz = {0, 0, 0, 0};
#if __clang_major__ >= 23
        const v8i gz8 = {0, 0, 0, 0, 0, 0, 0, 0};
        __builtin_amdgcn_tensor_load_to_lds(g0, g1, gz, gz, gz8, 0);
#else
        __builtin_amdgcn_tensor_load_to_lds(g0, g1, gz, gz, 0);
#endif
        __builtin_amdgcn_s_wait_tensorcnt(0);
    }
#else
    for (int e = tid; e < NN * NN; e += 512) {
        const int i = e >> 8, j = e & 255;
        P[i * PITCH + j] = Kb[e];
    }
#endif

    // A-matrix 16x4 f32 layout (ISA table): lane L -> M = L%16,
    // K = {0,1} (lanes 0-15) or {2,3} (lanes 16-31) across the 2 VGPRs.
    const int kb = (lane >> 4) * 2;
    const int mr = lane & 15;

    for (int it = 0; it < 10; ++it) {
        __syncthreads();
        // ---- row sums: rowsum = P_tile x ones  (wave w -> rows 16w..16w+15)
        {
            const int r0 = wave * 16;
            v8f c = {0.f, 0.f, 0.f, 0.f, 0.f, 0.f, 0.f, 0.f};
            v2f one; one.x = 1.0f; one.y = 1.0f;
            const float* rowp = &P[(r0 + mr) * PITCH + kb];
            for (int kt = 0; kt < 64; ++kt) {
                v2f a; a.x = rowp[kt * 4]; a.y = rowp[kt * 4 + 1];
                c = __builtin_amdgcn_wmma_f32_16x16x4_f32(
                        false, a, false, one, (short)0, c, false, false);
            }
            // D layout: vgpr r = row r (lanes 0-15) / row r+8 (lanes 16-31)
            if (lane == 0) {
                #pragma unroll
                for (int r = 0; r < 8; ++r) rs[r0 + r] = c[r];
            } else if (lane == 16) {
                #pragma unroll
                for (int r = 0; r < 8; ++r) rs[r0 + 8 + r] = c[r];
            }
        }
        __syncthreads();
        if (tid < NN) rs[tid] = 1.0f / fmaxf(rs[tid], 1e-8f);  // 256 divides, not 65536
        __syncthreads();
        for (int e = tid; e < NN * NN; e += 512) {
            const int i = e >> 8, j = e & 255;
            P[i * PITCH + j] *= rs[i];
        }
        __syncthreads();
        // ---- col sums: colsum = ones x P_tile  (wave w -> cols 16w..16w+15)
        {
            const int c0 = wave * 16;
            v8f c = {0.f, 0.f, 0.f, 0.f, 0.f, 0.f, 0.f, 0.f};
            v2f one; one.x = 1.0f; one.y = 1.0f;   // all-ones A => layout-robust
            const int n = lane & 15;
            const int krow = (lane >> 4) * 2;
            for (int kt = 0; kt < 64; ++kt) {
                v2f q;
                q.x = P[(kt * 4 + krow)     * PITCH + c0 + n];
                q.y = P[(kt * 4 + krow + 1) * PITCH + c0 + n];
                c = __builtin_amdgcn_wmma_f32_16x16x4_f32(
                        false, one, false, q, (short)0, c, false, false);
            }
            if (lane < 16) cs[c0 + lane] = c[0];   // D[0, n] = colsum[c0+n]
        }
        __syncthreads();
        if (tid < NN) cs[tid] = 1.0f / fmaxf(cs[tid], 1e-8f);
        __syncthreads();
        for (int e = tid; e < NN * NN; e += 512) {
            const int i = e >> 8, j = e & 255;
            P[i * PITCH + j] *= cs[j];
        }
    }
    __syncthreads();
    // ---- loss = sum(P * W)
    float acc = 0.f;
    for (int e = tid; e < NN * NN; e += 512) {
        const int i = e >> 8, j = e & 255;
        acc += P[i * PITCH + j] * Wb[e];
    }
    acc = waveReduceSum(acc);
    __syncthreads();
    if (lane == 0) cs[wave] = acc;
    __syncthreads();
    if (tid == 0) {
        float s = 0.f;
        for (int wv = 0; wv < 16; ++wv) s += cs[wv];
        losses[b] = s;
    }
}

__global__ void finish_loss(const float* __restrict__ losses,
                            float* __restrict__ out) {
    if (threadIdx.x == 0 && blockIdx.x == 0) {
        const float s = losses[0] + losses[1] + losses[2] + losses[3];
        out[0] = 0.001f * 0.25f * s;   // LOSS_SCALE * mean over B=4
    }
}

// ---------------------------------------------------------------------------
extern "C" void kernel_launch(void* const* d_in, const int* in_sizes, int n_in,
                              void* d_out, int out_size, void* d_ws, size_t ws_size,
                              hipStream_t stream) {
    const float* y_s = (const float*)d_in[0];
    const float* y_t = (const float*)d_in[1];
    float* ws  = (float*)d_ws;
    float* p_s = ws;                         // 4*256*512
    float* p_t = p_s + NB * NN * ND;         // 4*256*512
    float* Wg  = p_t + NB * NN * ND;         // 4*256*256
    float* Kg  = Wg + NB * NN * NN;          // 4*256*256
    float* losses = Kg + NB * NN * NN;       // 4

    softmax_rows<<<2 * NB * NN / 8, 256, 0, stream>>>(y_s, y_t, p_s, p_t);

    dim3 cgrid(NN / 16, NN / 16, NB);
    cost_matrix<<<cgrid, 256, 0, stream>>>(p_s, p_t, Wg, Kg);

    const size_t lds_bytes = (size_t)(NN * PITCH + 2 * NN) * sizeof(float); // ~265 KB
    sinkhorn_lds<<<NB, 512, lds_bytes, stream>>>(Wg, Kg, losses);

    finish_loss<<<1, 32, 0, stream>>>(losses, (float*)d_out);
}